// ProbEncoder_59219009077524
// MI455X (gfx1250) — compile-verified
//
#include <hip/hip_runtime.h>
#include <hip/hip_bf16.h>

// Problem constants (reference: B=4, L=512, D=C=32, K=5 buckets).
#define BATCH 4
#define LEN   512
#define DD    32
#define NROW  (BATCH * LEN)      // 2048
#define KDIM  160                // 5 buckets * 32 b-channels

typedef float v2f __attribute__((ext_vector_type(2)));
typedef float v8f __attribute__((ext_vector_type(8)));

// ---------------------------------------------------------------------------
// Kernel 1: W[a][k*32+b] = sum_c (ternary[0,k,a,b,c] + ternary[1,k,a,b,c])
// ternary layout: [2][5][32(a)][32(b)][32(c)]
// ---------------------------------------------------------------------------
__global__ void reduce_W_kernel(const float* __restrict__ ternary,
                                float* __restrict__ W) {
    int t = blockIdx.x * blockDim.x + threadIdx.x;   // 0..5119
    if (t >= DD * KDIM) return;
    int a  = t / KDIM;
    int kb = t % KDIM;
    int k = kb >> 5, b = kb & 31;
    const float* p0 = ternary + ((((size_t)k      ) * 32 + a) * 32 + b) * 32;
    const float* p1 = ternary + ((((size_t)k + 5  ) * 32 + a) * 32 + b) * 32;
    float s = 0.f;
    #pragma unroll
    for (int c = 0; c < 32; ++c) s += p0[c] + p1[c];
    W[t] = s;                                        // t == a*KDIM + kb
}

// ---------------------------------------------------------------------------
// Kernel 2: u[row][b] = valid(row) * softmax(x[row,:])[b]
// ---------------------------------------------------------------------------
__global__ void softmax_u_kernel(const float* __restrict__ x,
                                 const int* __restrict__ mask,
                                 float* __restrict__ u) {
    int row = blockIdx.x * blockDim.x + threadIdx.x; // 0..2047
    if (row >= NROW) return;
    const float* xr = x + (size_t)row * DD;
    float v[DD];
    float m = -__builtin_inff();
    #pragma unroll
    for (int b = 0; b < DD; ++b) { v[b] = xr[b]; m = fmaxf(m, v[b]); }
    float s = 0.f;
    #pragma unroll
    for (int b = 0; b < DD; ++b) { v[b] = expf(v[b] - m); s += v[b]; }
    float inv  = 1.0f / s;
    float keep = (mask[row] != 0) ? 1.0f : 0.0f;
    float* ur = u + (size_t)row * DD;
    #pragma unroll
    for (int b = 0; b < DD; ++b) ur[b] = keep * v[b] * inv;
}

// ---------------------------------------------------------------------------
// Kernel 3: inclusive prefix scan along j for each (z, b):
// P[z,j,b] = sum_{j'<=j} u[z,j',b].  One workgroup per z, 512 threads,
// 3-phase scan with 2KB LDS of segment sums.
// ---------------------------------------------------------------------------
__global__ void scan_P_kernel(const float* __restrict__ u,
                              float* __restrict__ P) {
    __shared__ float seg[16 * 32];
    int z = blockIdx.x;                 // 0..3
    int t = threadIdx.x;                // 0..511
    int b = t & 31, s = t >> 5;         // column, segment (16 segs of 32 rows)
    const float* uz = u + (size_t)z * LEN * DD;
    float* Pz       = P + (size_t)z * LEN * DD;

    float sum = 0.f;
    for (int j = s * 32; j < s * 32 + 32; ++j) sum += uz[j * DD + b];
    seg[s * 32 + b] = sum;
    __syncthreads();
    if (t < 32) {
        float run = 0.f;
        #pragma unroll
        for (int s2 = 0; s2 < 16; ++s2) {
            float tmp = seg[s2 * 32 + t];
            seg[s2 * 32 + t] = run;     // exclusive segment offset
            run += tmp;
        }
    }
    __syncthreads();
    float run = seg[s * 32 + b];
    for (int j = s * 32; j < s * 32 + 32; ++j) {
        run += uz[j * DD + b];
        Pz[j * DD + b] = run;
    }
}

// ---------------------------------------------------------------------------
// Kernel 4: assemble GEMM A-matrix S[row][160] with the 5 bucket sums,
// scaled by valid(row)/32 (the 1/C of uniform q_y, plus the i-mask).
//   kb [0,32):   prefix  sum_{j<=i-3} u[j]
//   kb [32,64):  u[i-2]
//   kb [64,96):  u[i-1]
//   kb [96,128): u[i+1] + u[i+2]
//   kb [128,160): suffix sum_{j>=i+3} u[j] = total - P[i+2]
// ---------------------------------------------------------------------------
__global__ void build_S_kernel(const float* __restrict__ u,
                               const float* __restrict__ P,
                               const int* __restrict__ mask,
                               float* __restrict__ S) {
    int row = blockIdx.x * blockDim.x + threadIdx.x; // 0..2047
    if (row >= NROW) return;
    int z = row >> 9, i = row & (LEN - 1);
    const float* uz = u + (size_t)z * LEN * DD;
    const float* Pz = P + (size_t)z * LEN * DD;
    float scale = (mask[row] != 0) ? (1.0f / 32.0f) : 0.0f;
    float* Sr = S + (size_t)row * KDIM;
    #pragma unroll
    for (int b = 0; b < DD; ++b) {
        float s0 = (i >= 3) ? Pz[(i - 3) * DD + b] : 0.f;
        float s1 = (i >= 2) ? uz[(i - 2) * DD + b] : 0.f;
        float s2 = (i >= 1) ? uz[(i - 1) * DD + b] : 0.f;
        float s3 = ((i + 1 < LEN) ? uz[(i + 1) * DD + b] : 0.f)
                 + ((i + 2 < LEN) ? uz[(i + 2) * DD + b] : 0.f);
        float s4 = (i + 2 < LEN)
                 ? (Pz[(LEN - 1) * DD + b] - Pz[(i + 2) * DD + b]) : 0.f;
        Sr[        b] = scale * s0;
        Sr[ 32  +  b] = scale * s1;
        Sr[ 64  +  b] = scale * s2;
        Sr[ 96  +  b] = scale * s3;
        Sr[128  +  b] = scale * s4;
    }
}

// ---------------------------------------------------------------------------
// Kernel 5: out[2048,32] = x + S[2048,160] @ W^T via V_WMMA_F32_16X16X4_F32.
// One wave per 16x16 output tile; 128 M-tiles x 2 N-tiles = 256 waves.
// A frag (16x4 f32): lane -> M = lane%16, vgpr v -> K = v + 2*(lane/16).
// B frag (4x16 f32): lane -> N = lane%16, vgpr v -> K = v + 2*(lane/16).
// C/D   (16x16 f32): vgpr r -> M = r + 8*(lane/16), N = lane%16.
// W is stored [a][kb] so both fragments are contiguous float2 loads.
// ---------------------------------------------------------------------------
__global__ void gemm_wmma_kernel(const float* __restrict__ S,
                                 const float* __restrict__ W,
                                 const float* __restrict__ x,
                                 float* __restrict__ out) {
    const int lane = threadIdx.x & 31;
    const int wave = threadIdx.x >> 5;
    const int tile = blockIdx.x * (blockDim.x >> 5) + wave;  // 0..255
    const int mt = tile >> 1;
    const int nt = tile & 1;
    const int row0 = mt * 16;
    const int a0   = nt * 16;
    const int lo = lane & 15;
    const int hi = lane >> 4;

    const float* Arow = S + (size_t)(row0 + lo) * KDIM + 2 * hi;
    const float* Brow = W + (size_t)(a0   + lo) * KDIM + 2 * hi;

    v8f c = {0.f, 0.f, 0.f, 0.f, 0.f, 0.f, 0.f, 0.f};
    #pragma unroll
    for (int kb = 0; kb < KDIM; kb += 4) {
        v2f a = *(const v2f*)(Arow + kb);
        v2f b = *(const v2f*)(Brow + kb);
        c = __builtin_amdgcn_wmma_f32_16x16x4_f32(
                /*neg_a=*/false, a, /*neg_b=*/false, b,
                /*c_mod=*/(short)0, c, /*reuse_a=*/false, /*reuse_b=*/false);
    }
    #pragma unroll
    for (int r = 0; r < 8; ++r) {
        size_t idx = (size_t)(row0 + r + 8 * hi) * DD + (a0 + lo);
        out[idx] = x[idx] + c[r];
    }
}

// ---------------------------------------------------------------------------
extern "C" void kernel_launch(void* const* d_in, const int* in_sizes, int n_in,
                              void* d_out, int out_size, void* d_ws, size_t ws_size,
                              hipStream_t stream) {
    const float* x      = (const float*)d_in[0];   // [4,512,32]
    const int*   mask   = (const int*)d_in[1];     // [4,512]
    const float* ternary = (const float*)d_in[2];  // [2,5,32,32,32]
    float* out = (float*)d_out;                    // [4,512,32]

    float* ws = (float*)d_ws;
    float* W = ws;                 //  5,120 f32  : [32(a)][160(kb)]
    float* u = W + DD * KDIM;      // 65,536 f32  : [2048][32]
    float* P = u + NROW * DD;      // 65,536 f32  : [2048][32]
    float* S = P + NROW * DD;      // 327,680 f32 : [2048][160]

    reduce_W_kernel <<<(DD * KDIM + 255) / 256, 256, 0, stream>>>(ternary, W);
    softmax_u_kernel<<<(NROW + 255) / 256,      256, 0, stream>>>(x, mask, u);
    scan_P_kernel   <<<BATCH,                   512, 0, stream>>>(u, P);
    build_S_kernel  <<<(NROW + 255) / 256,      256, 0, stream>>>(u, P, mask, S);
    gemm_wmma_kernel<<<32,                      256, 0, stream>>>(S, W, x, out);
}